// TorsoGCNv2_78168404787866
// MI455X (gfx1250) — compile-verified
//
#include <hip/hip_runtime.h>

typedef float v2f __attribute__((ext_vector_type(2)));
typedef float v8f __attribute__((ext_vector_type(8)));

#define N_NODES  50000
#define N_EDGES  800000
#define N_GRAPHS 64
#define OUT_DIM  768

// ---------------- utility kernels ----------------

__global__ void zero_kernel(float* __restrict__ p, long long n) {
    long long i = (long long)blockIdx.x * blockDim.x + threadIdx.x;
    long long stride = (long long)gridDim.x * blockDim.x;
    for (; i < n; i += stride) p[i] = 0.0f;
}

__global__ void max_kernel(const float* __restrict__ a, unsigned* __restrict__ out, int n) {
    __shared__ float red[256];
    int tid = threadIdx.x;
    long long i = (long long)blockIdx.x * blockDim.x + tid;
    long long stride = (long long)gridDim.x * blockDim.x;
    float m = 0.0f;
    for (; i < n; i += stride) m = fmaxf(m, a[i]);
    red[tid] = m;
    __syncthreads();
    for (int s = 128; s > 0; s >>= 1) {
        if (tid < s) red[tid] = fmaxf(red[tid], red[tid + s]);
        __syncthreads();
    }
    if (tid == 0) atomicMax(out, __float_as_uint(red[0]));   // weights >= 0
}

__global__ void deg_kernel(const float* __restrict__ attr, const int* __restrict__ dst,
                           float* __restrict__ deg, int n) {
    int e = blockIdx.x * blockDim.x + threadIdx.x;
    if (e < n) atomicAdd(&deg[dst[e]], attr[e]);   // unscaled; scale applied later
}

__global__ void dinv_kernel(const float* __restrict__ deg, const unsigned* __restrict__ maxbits,
                            float* __restrict__ invdeg, float* __restrict__ dinv, int n) {
    int i = blockIdx.x * blockDim.x + threadIdx.x;
    if (i >= n) return;
    float inv_max = 1.0f / __uint_as_float(*maxbits);
    float d = deg[i] * inv_max + 1.0f;            // + self loop
    invdeg[i] = 1.0f / d;
    dinv[i]   = rsqrtf(d);
}

__global__ void norm_kernel(const float* __restrict__ attr, const int* __restrict__ src,
                            const int* __restrict__ dst, const float* __restrict__ dinv,
                            const unsigned* __restrict__ maxbits, float* __restrict__ nrm, int n) {
    int e = blockIdx.x * blockDim.x + threadIdx.x;
    if (e >= n) return;
    float inv_max = 1.0f / __uint_as_float(*maxbits);
    nrm[e] = dinv[src[e]] * (attr[e] * inv_max) * dinv[dst[e]];
}

// ------- WMMA f32 GEMM: C[M,N] = A[M,K] @ B[K,N] (+bias) -------
// One wave computes a 16x(16*NT) strip of C via V_WMMA_F32_16X16X4_F32:
// A fragment loaded once per k-step, reused by NT WMMAs into NT accumulators
// (no D->A/B hazards between them).
// A-frag layout (ISA 7.12.2, 16x4 f32): lane L<16 -> row m0+L, K pair {k0,k0+1};
// lanes 16..31 -> same rows, K pair {k0+2,k0+3}.  B-frag mirrors over N.
template <int NT>
__global__ void gemm_wmma(const float* __restrict__ A, const float* __restrict__ B,
                          const float* __restrict__ bias, float* __restrict__ C,
                          int M, int N, int K) {
    int wave = (int)((blockIdx.x * blockDim.x + threadIdx.x) >> 5);
    int lane = threadIdx.x & 31;
    int stripsN = N / (16 * NT);
    int strips  = (M >> 4) * stripsN;
    if (wave >= strips) return;                   // uniform per wave: EXEC stays all-ones
    int m0 = (wave / stripsN) << 4;
    int n0 = (wave % stripsN) * (16 * NT);
    int hl = lane >> 4;                           // 0 or 1: selects K sub-pair
    int ll = lane & 15;

    const float* arow  = A + (size_t)(m0 + ll) * K + 2 * hl;
    const float* bbase = B + (size_t)(2 * hl) * N + (n0 + ll);

    v8f acc[NT];
    #pragma unroll
    for (int t = 0; t < NT; ++t) acc[t] = (v8f){};

    for (int k0 = 0; k0 < K; k0 += 4) {
        v2f a;
        a.x = arow[k0];
        a.y = arow[k0 + 1];
        #pragma unroll
        for (int t = 0; t < NT; ++t) {
            const float* bc = bbase + 16 * t;
            v2f b;
            b.x = bc[(size_t)k0 * N];
            b.y = bc[(size_t)(k0 + 1) * N];
            acc[t] = __builtin_amdgcn_wmma_f32_16x16x4_f32(
                /*neg_a=*/false, a, /*neg_b=*/false, b,
                /*c_mod=*/(short)0, acc[t], /*reuse_a=*/false, /*reuse_b=*/false);
        }
    }

    #pragma unroll
    for (int t = 0; t < NT; ++t) {
        if (bias) {
            float bv = bias[n0 + 16 * t + ll];
            #pragma unroll
            for (int v = 0; v < 8; ++v) acc[t][v] += bv;
        }
        // C/D layout: VGPR v, lane half hl -> row m0 + 8*hl + v, col n0 + 16*t + ll
        float* crow = C + (size_t)(m0 + 8 * hl) * N + (n0 + 16 * t + ll);
        #pragma unroll
        for (int v = 0; v < 8; ++v) crow[(size_t)v * N] = acc[t][v];
    }
}

// ---------------- edge scatter-aggregate ----------------
// thread t -> edge e = t>>shift, feature quad f4 = (t & mask)*4.
// float4 gather of h[src]; 4 atomics land in the same L2 cacheline of dst row.
// shift = log2(F/4).
__global__ void edge_agg(const float* __restrict__ h, const float* __restrict__ nrm,
                         const int* __restrict__ src, const int* __restrict__ dst,
                         float* __restrict__ agg, int shift) {
    long long t = (long long)blockIdx.x * blockDim.x + threadIdx.x;
    long long total = (long long)N_EDGES << shift;
    if (t >= total) return;
    int e  = (int)(t >> shift);
    int f4 = (int)(t & ((1 << shift) - 1)) << 2;
    long long F = 4LL << shift;
    float4 hv = *(const float4*)(h + (long long)src[e] * F + f4);
    float w = nrm[e];
    float* ap = agg + (long long)dst[e] * F + f4;
    atomicAdd(ap + 0, hv.x * w);
    atomicAdd(ap + 1, hv.y * w);
    atomicAdd(ap + 2, hv.z * w);
    atomicAdd(ap + 3, hv.w * w);
}

// out = relu(agg + h*invdeg + b), in place over agg (float4). shift = log2(F/4).
__global__ void post_kernel(float4* __restrict__ agg, const float4* __restrict__ h,
                            const float* __restrict__ invdeg, const float* __restrict__ b,
                            int shift) {
    long long t = (long long)blockIdx.x * blockDim.x + threadIdx.x;
    long long total = (long long)N_NODES << shift;
    if (t >= total) return;
    int i  = (int)(t >> shift);
    int f4 = (int)(t & ((1 << shift) - 1)) << 2;
    float id = invdeg[i];
    float4 a = agg[t];
    float4 hh = h[t];
    a.x = fmaxf(a.x + hh.x * id + b[f4 + 0], 0.0f);
    a.y = fmaxf(a.y + hh.y * id + b[f4 + 1], 0.0f);
    a.z = fmaxf(a.z + hh.z * id + b[f4 + 2], 0.0f);
    a.w = fmaxf(a.w + hh.w * id + b[f4 + 3], 0.0f);
    agg[t] = a;
}

// ---------------- pooling ----------------
__global__ void pool_kernel(const float* __restrict__ h3, const int* __restrict__ batch,
                            float* __restrict__ pooled, float* __restrict__ cnt) {
    long long t = (long long)blockIdx.x * blockDim.x + threadIdx.x;
    long long total = (long long)N_NODES * 8;     // 32 features / 4 per thread
    if (t >= total) return;
    int i  = (int)(t >> 3);
    int f4 = (int)(t & 7) << 2;
    int g = batch[i];
    float4 hv = *(const float4*)(h3 + (long long)i * 32 + f4);
    float* pp = pooled + g * 32 + f4;
    atomicAdd(pp + 0, hv.x);
    atomicAdd(pp + 1, hv.y);
    atomicAdd(pp + 2, hv.z);
    atomicAdd(pp + 3, hv.w);
    if ((t & 7) == 0) atomicAdd(&cnt[g], 1.0f);
}

__global__ void pdiv_kernel(float* __restrict__ pooled, const float* __restrict__ cnt) {
    int t = blockIdx.x * blockDim.x + threadIdx.x;
    if (t < N_GRAPHS * 32) pooled[t] /= fmaxf(cnt[t >> 5], 1.0f);
}

// ---------------- launcher ----------------

extern "C" void kernel_launch(void* const* d_in, const int* in_sizes, int n_in,
                              void* d_out, int out_size, void* d_ws, size_t ws_size,
                              hipStream_t stream) {
    const float* x    = (const float*)d_in[0];
    const int*   ei   = (const int*)d_in[1];
    const int*   src  = ei;
    const int*   dst  = ei + N_EDGES;
    const float* attr = (const float*)d_in[2];
    const int*   batch= (const int*)d_in[3];
    const float* W1   = (const float*)d_in[4];
    const float* b1   = (const float*)d_in[5];
    const float* W2   = (const float*)d_in[6];
    const float* b2   = (const float*)d_in[7];
    const float* W3   = (const float*)d_in[8];
    const float* b3   = (const float*)d_in[9];
    const float* Wl   = (const float*)d_in[10];
    const float* bl   = (const float*)d_in[11];
    float* out = (float*)d_out;

    // workspace layout (floats)
    float* ws      = (float*)d_ws;
    float* buf_h   = ws;                                  // 50000*128
    float* buf_a   = buf_h  + (long long)N_NODES * 128;   // 50000*128
    float* deg     = buf_a  + (long long)N_NODES * 128;   // 50000
    float* invdeg  = deg    + N_NODES;                    // 50000
    float* dinv    = invdeg + N_NODES;                    // 50000
    float* nrm     = dinv   + N_NODES;                    // 800000
    float* pooled  = nrm    + N_EDGES;                    // 64*32
    float* cnt     = pooled + N_GRAPHS * 32;              // 64
    unsigned* maxb = (unsigned*)(cnt + N_GRAPHS);         // 1

    auto cdiv = [](long long a, long long b) -> unsigned { return (unsigned)((a + b - 1) / b); };

    // 1) zero deg..maxb region (invdeg/dinv/nrm get fully overwritten anyway)
    long long zn = 3LL * N_NODES + N_EDGES + N_GRAPHS * 32 + N_GRAPHS + 1;
    zero_kernel<<<1024, 256, 0, stream>>>(deg, zn);

    // 2) max(edge_attr)
    max_kernel<<<512, 256, 0, stream>>>(attr, maxb, N_EDGES);

    // 3) degree accumulation, node scales, per-edge norm (reused by all 3 layers)
    deg_kernel<<<cdiv(N_EDGES, 256), 256, 0, stream>>>(attr, dst, deg, N_EDGES);
    dinv_kernel<<<cdiv(N_NODES, 256), 256, 0, stream>>>(deg, maxb, invdeg, dinv, N_NODES);
    norm_kernel<<<cdiv(N_EDGES, 256), 256, 0, stream>>>(attr, src, dst, dinv, maxb, nrm, N_EDGES);

    // ---- layer 1: 128 -> 128 ----
    {
        long long waves = (50000 / 16) * (128 / 64);      // NT=4 strips
        gemm_wmma<4><<<cdiv(waves * 32, 256), 256, 0, stream>>>(x, W1, nullptr, buf_h, N_NODES, 128, 128);
        zero_kernel<<<2048, 256, 0, stream>>>(buf_a, (long long)N_NODES * 128);
        edge_agg<<<cdiv((long long)N_EDGES << 5, 256), 256, 0, stream>>>(buf_h, nrm, src, dst, buf_a, 5);
        post_kernel<<<cdiv((long long)N_NODES << 5, 256), 256, 0, stream>>>(
            (float4*)buf_a, (const float4*)buf_h, invdeg, b1, 5);
    }
    // ---- layer 2: 128 -> 64 ----
    {
        long long waves = (50000 / 16) * (64 / 64);       // NT=4
        gemm_wmma<4><<<cdiv(waves * 32, 256), 256, 0, stream>>>(buf_a, W2, nullptr, buf_h, N_NODES, 64, 128);
        zero_kernel<<<2048, 256, 0, stream>>>(buf_a, (long long)N_NODES * 64);
        edge_agg<<<cdiv((long long)N_EDGES << 4, 256), 256, 0, stream>>>(buf_h, nrm, src, dst, buf_a, 4);
        post_kernel<<<cdiv((long long)N_NODES << 4, 256), 256, 0, stream>>>(
            (float4*)buf_a, (const float4*)buf_h, invdeg, b2, 4);
    }
    // ---- layer 3: 64 -> 32 ----
    {
        long long waves = (50000 / 16) * (32 / 32);       // NT=2
        gemm_wmma<2><<<cdiv(waves * 32, 256), 256, 0, stream>>>(buf_a, W3, nullptr, buf_h, N_NODES, 32, 64);
        zero_kernel<<<2048, 256, 0, stream>>>(buf_a, (long long)N_NODES * 32);
        edge_agg<<<cdiv((long long)N_EDGES << 3, 256), 256, 0, stream>>>(buf_h, nrm, src, dst, buf_a, 3);
        post_kernel<<<cdiv((long long)N_NODES << 3, 256), 256, 0, stream>>>(
            (float4*)buf_a, (const float4*)buf_h, invdeg, b3, 3);
    }

    // ---- global mean pool ----
    pool_kernel<<<cdiv((long long)N_NODES * 8, 256), 256, 0, stream>>>(buf_a, batch, pooled, cnt);
    pdiv_kernel<<<8, 256, 0, stream>>>(pooled, cnt);

    // ---- final linear: [64,32] @ [32,768] + bl -> d_out [64,768] ----
    {
        long long waves = (64 / 16) * (768 / 64);         // NT=4
        gemm_wmma<4><<<cdiv(waves * 32, 256), 256, 0, stream>>>(pooled, Wl, bl, out, 64, 768, 32);
    }
}